// SubbandCompose_39728447488307
// MI455X (gfx1250) — compile-verified
//
#include <hip/hip_runtime.h>
#include <hip/hip_bf16.h>

// SubbandCompose for MI455X (gfx1250, wave32, WMMA).
//
// Math: irfft(65-pt real spectrum, norm='forward') == y = x @ M^T with
//   M[t,k] = w_k * cos(pi*k*t/64), w_0 = w_64 = 1, else 2   (t=0..127)
// followed by a 6-tap polyphase overlap-add with filter, scaled by 64/32768.
//
// Strategy: per 256-thread block, DCT-GEMM a 128-row tile into LDS via
// V_WMMA_F32_16X16X4_F32 (8 waves x 16 rows x 128 cols, K padded 65->68),
// then overlap-add 112 output rows from LDS (5-row halo stays in-tile).

typedef float v2f __attribute__((ext_vector_type(2)));
typedef float v8f __attribute__((ext_vector_type(8)));

#define WIN        64
#define NFREQ      65
#define KPAD       68          // 65 padded to multiple of 4
#define NDCT       128         // irfft output length
#define ROWS_TILE  128         // DCT rows computed per block (8 waves x 16)
#define TB         112         // output time-steps emitted per block (halo=5, pad to 117<=128)
#define T_IN       8192
#define NBATCH     16
#define TRUE_LEN   (T_IN - 5)  // 8187
#define TOTAL_ROWS (NBATCH * T_IN)
#define BPB        ((TRUE_LEN + TB - 1) / TB)   // 74 blocks per batch
#define OUT_SCALE  (64.0f / 32768.0f)

// LDS byte offsets (dynamic shared memory, ~133.5 KB total)
#define XS_BYTES   (ROWS_TILE * KPAD * 4)            // 34816
#define BS_BYTES   ((KPAD / 2) * NDCT * 8)           // 34816 (float2 per K-pair)
#define YS_BYTES   (ROWS_TILE * NDCT * 4)            // 65536
#define FS_BYTES   (6 * WIN * 4)                     // 1536
#define SMEM_BYTES (XS_BYTES + BS_BYTES + YS_BYTES + FS_BYTES)

__device__ __forceinline__ float dct_entry(int k, int n) {
    // B[k][n] = M[n][k]; zero for padded K
    if (k > WIN) return 0.0f;
    float wk = (k == 0 || k == WIN) ? 1.0f : 2.0f;
    float ang = (float)(k * n) * (3.14159265358979323846f / 64.0f);
    return wk * __cosf(ang);
}

__global__ void __launch_bounds__(256)
subband_compose_kernel(const float* __restrict__ inp,
                       const float* __restrict__ filt,
                       float* __restrict__ out) {
    extern __shared__ char smem[];
    float* Xs = (float*)smem;                              // [128][68]
    v2f*   Bs = (v2f*)(smem + XS_BYTES);                   // [34][128] K-pairs
    float* Ys = (float*)(smem + XS_BYTES + BS_BYTES);      // [128][128]
    float* Fs = (float*)(smem + XS_BYTES + BS_BYTES + YS_BYTES); // [384]

    const int tid = threadIdx.x;
    const int s   = blockIdx.x / BPB;       // batch
    const int bb  = blockIdx.x % BPB;
    const int t0  = bb * TB;                // first output time-step

    // ---- Stage 0: filter (pre-scaled) ----
    for (int i = tid; i < 6 * WIN; i += 256)
        Fs[i] = filt[i] * OUT_SCALE;

    // ---- Stage 1: input tile -> LDS, K padded with zeros ----
    {
        const float* base = inp;
        __builtin_prefetch(&base[(size_t)(s * T_IN + t0) * NFREQ], 0, 1);
        for (int i = tid; i < ROWS_TILE * KPAD; i += 256) {
            int r = i / KPAD, f = i - r * KPAD;
            int gr = s * T_IN + t0 + r;
            if (gr > TOTAL_ROWS - 1) gr = TOTAL_ROWS - 1;   // clamp (unused rows)
            Xs[i] = (f < NFREQ) ? base[(size_t)gr * NFREQ + f] : 0.0f;
        }
    }

    // ---- Stage 2: cosine matrix as K-pairs: Bs[kp*128+n] = {B[2kp][n], B[2kp+1][n]} ----
    for (int i = tid; i < (KPAD / 2) * NDCT; i += 256) {
        int kp = i / NDCT, n = i - kp * NDCT;
        v2f b;
        b.x = dct_entry(2 * kp,     n);
        b.y = dct_entry(2 * kp + 1, n);
        Bs[i] = b;
    }
    __syncthreads();

    // ---- Stage 3: DCT GEMM via V_WMMA_F32_16X16X4_F32 ----
    // Wave w computes rows [16w, 16w+16) x all 128 cols.
    {
        const int wave  = tid >> 5;
        const int lane  = tid & 31;
        const int lr    = lane & 15;          // row-in-tile / col-in-tile
        const int khalf = (lane >> 4) << 1;   // lanes 0-15: K+{0,1}; 16-31: K+{2,3}
        const int m0    = wave << 4;

        v8f acc[8];
        #pragma unroll
        for (int nt = 0; nt < 8; ++nt) acc[nt] = (v8f)(0.0f);

        const float* xrow = &Xs[(m0 + lr) * KPAD];
        for (int ks = 0; ks < KPAD / 4; ++ks) {   // 17 K-steps of 4
            const int ka = (ks << 2) + khalf;     // even
            v2f a = *(const v2f*)(xrow + ka);     // A frag: X[row][ka..ka+1]
            const v2f* brow = &Bs[(ka >> 1) * NDCT];
            #pragma unroll
            for (int nt = 0; nt < 8; ++nt) {
                v2f b = brow[(nt << 4) + lr];     // B frag: {B[ka][n], B[ka+1][n]}
                acc[nt] = __builtin_amdgcn_wmma_f32_16x16x4_f32(
                    false, a, false, b, (short)0, acc[nt], false, false);
            }
        }

        // D layout: VGPR v, lanes 0-15 -> M=v, lanes 16-31 -> M=v+8; N = lane&15
        const int moff = m0 + ((lane >> 4) << 3);
        #pragma unroll
        for (int nt = 0; nt < 8; ++nt)
            #pragma unroll
            for (int v = 0; v < 8; ++v)
                Ys[(moff + v) * NDCT + (nt << 4) + lr] = acc[nt][v];
    }
    __syncthreads();

    // ---- Stage 4: 6-tap polyphase overlap-add, stream to HBM ----
    {
        int tcount = TRUE_LEN - t0;
        if (tcount > TB) tcount = TB;
        for (int i = tid; i < tcount * WIN; i += 256) {
            int lt = i >> 6;            // local output time-step
            int j  = i & (WIN - 1);
            float v = 0.0f;
            #pragma unroll
            for (int k = 0; k < 6; ++k) {
                int r   = lt + 5 - k;
                int col = (k & 1) ? (WIN + j) : j;
                v += Fs[k * WIN + j] * Ys[r * NDCT + col];
            }
            int tp = t0 + lt;
            out[(size_t)s * (TRUE_LEN * WIN) + (size_t)tp * WIN + j] = v;
        }
    }
}

extern "C" void kernel_launch(void* const* d_in, const int* in_sizes, int n_in,
                              void* d_out, int out_size, void* d_ws, size_t ws_size,
                              hipStream_t stream) {
    const float* inp  = (const float*)d_in[0];   // (16,1,8192,65) f32
    const float* filt = (const float*)d_in[1];   // (384,) f32
    // d_in[2] = lengths (all 8192) -- constant in this problem size
    float* out = (float*)d_out;                  // (16,1,8187*64) f32

    dim3 grid(NBATCH * BPB);   // 16 * 74 = 1184 blocks
    dim3 block(256);
    subband_compose_kernel<<<grid, block, SMEM_BYTES, stream>>>(inp, filt, out);
}